// NFLinear_1219770712333
// MI455X (gfx1250) — compile-verified
//
#include <hip/hip_runtime.h>
#include <hip/hip_bf16.h>

typedef float v2f __attribute__((ext_vector_type(2)));
typedef float v8f __attribute__((ext_vector_type(8)));

// ---------------- workspace layout (float offsets) ----------------
#define OFF_U      0        // [512*16]  sum over a of w_ih      (scale 1/784)
#define OFF_RW     8192     // [784*16]  sum over b of w_ih      (scale 1/512)
#define OFF_DIAG   20736    // [512*16]  diag(w_hh)              (raw)
#define OFF_ROWW   28928    // [512*16]  sum over s of w_hh      (scale 1/512)
#define OFF_COLW   37120    // [512*16]  sum over r of w_hh      (scale 1/512)
#define OFF_MWIH   45312    // [16] mean(w_ih)
#define OFF_MALL   45328    // [16] mean(w_hh)
#define OFF_MDIAG  45344    // [16] mean(diag w_hh)
#define OFF_MB     45360    // [16] mean(b)
#define OFF_A0     45376    // [784*16] per-a additive for out0
#define OFF_B0     57920    // [512*16] per-b additive for out0
#define OFF_C0     66112    // [16]
#define OFF_ROW1   66128    // [512*16] per-r additive for out1
#define OFF_COL1   74320    // [512*16] per-s additive for out1
#define OFF_DIAG1  82512    // [512*16] diag-only additive for out1
#define OFF_C1     90704    // [16]
#define WS_FLOATS  90720

// ---------------- phase 1: pooled statistics ----------------

__global__ void k_zero(float* __restrict__ ws, int n) {
  int i = blockIdx.x * blockDim.x + threadIdx.x;
  if (i < n) ws[i] = 0.f;
}

// u[b,c] += sum over a-chunk of w_ih[a,b,c];  8 chunks of 98
__global__ void k_sum_u(const float* __restrict__ wih, float* __restrict__ u) {
  int tid = blockIdx.x * blockDim.x + threadIdx.x;   // 8*8192 threads
  int chunk = tid >> 13;
  int j = tid & 8191;
  const float* p = wih + (size_t)chunk * 98u * 8192u + j;
  float s = 0.f;
  #pragma unroll 7
  for (int a = 0; a < 98; ++a) s += p[(size_t)a * 8192u];
  atomicAdd(&u[j], s);
}

// rw[a,c] += sum over b-chunk of w_ih[a,b,c];  8 chunks of 64
__global__ void k_sum_rw(const float* __restrict__ wih, float* __restrict__ rw) {
  int tid = blockIdx.x * blockDim.x + threadIdx.x;
  if (tid >= 8 * 12544) return;
  int chunk = tid / 12544;
  int j = tid - chunk * 12544;
  int a = j >> 4, c = j & 15;
  const float* p = wih + (size_t)a * 8192u + (size_t)(chunk * 64) * 16u + c;
  float s = 0.f;
  #pragma unroll 8
  for (int b = 0; b < 64; ++b) s += p[b * 16];
  atomicAdd(&rw[j], s);
}

__global__ void k_diag(const float* __restrict__ whh, float* __restrict__ dg) {
  int tid = blockIdx.x * blockDim.x + threadIdx.x;   // 8192
  if (tid >= 8192) return;
  int r = tid >> 4, c = tid & 15;
  dg[tid] = whh[(size_t)r * 8192u + r * 16 + c];
}

// roww[r,c] += sum over s-chunk of w_hh[r,s,c]
__global__ void k_roww(const float* __restrict__ whh, float* __restrict__ rowv) {
  int tid = blockIdx.x * blockDim.x + threadIdx.x;   // 8*8192
  int chunk = tid >> 13;
  int j = tid & 8191;
  int r = j >> 4, c = j & 15;
  const float* p = whh + (size_t)r * 8192u + (size_t)(chunk * 64) * 16u + c;
  float s = 0.f;
  #pragma unroll 8
  for (int ss = 0; ss < 64; ++ss) s += p[ss * 16];
  atomicAdd(&rowv[j], s);
}

// colw[s,c] += sum over r-chunk of w_hh[r,s,c]
__global__ void k_colw(const float* __restrict__ whh, float* __restrict__ colv) {
  int tid = blockIdx.x * blockDim.x + threadIdx.x;   // 8*8192
  int chunk = tid >> 13;
  int j = tid & 8191;
  const float* p = whh + (size_t)(chunk * 64) * 8192u + j;
  float s = 0.f;
  #pragma unroll 8
  for (int r = 0; r < 64; ++r) s += p[(size_t)r * 8192u];
  atomicAdd(&colv[j], s);
}

__global__ void k_small(float* __restrict__ ws, const float* __restrict__ bvec) {
  int c = threadIdx.x;
  if (c >= 16) return;
  float s = 0.f;
  for (int b = 0; b < 512; ++b) s += ws[OFF_U + b * 16 + c];
  ws[OFF_MWIH + c] = s * (1.f / (784.f * 512.f));
  s = 0.f;
  for (int r = 0; r < 512; ++r) s += ws[OFF_ROWW + r * 16 + c];
  ws[OFF_MALL + c] = s * (1.f / (512.f * 512.f));
  s = 0.f;
  for (int r = 0; r < 512; ++r) s += ws[OFF_DIAG + r * 16 + c];
  ws[OFF_MDIAG + c] = s * (1.f / 512.f);
  s = 0.f;
  for (int r = 0; r < 512; ++r) s += bvec[r * 16 + c];
  ws[OFF_MB + c] = s * (1.f / 512.f);
}

// ---------------- phase 2: fold stats x theta into additive vectors ----------------

__global__ void k_combine(float* __restrict__ ws,
                          const float* __restrict__ bvec,
                          const float* __restrict__ T0, const float* __restrict__ bias0,
                          const float* __restrict__ T1, const float* __restrict__ bias1,
                          const float* __restrict__ T2, const float* __restrict__ bias2,
                          float* __restrict__ out2) {
  int tid = blockIdx.x * blockDim.x + threadIdx.x;
  if (tid >= 784 * 16) return;
  int i = tid >> 4, co = tid & 15;
  const float inv784 = 1.f / 784.f, inv512 = 1.f / 512.f;

  // A0[a] = rowmean(w_ih)[a] @ T0[1]
  {
    float s = 0.f;
    #pragma unroll
    for (int ci = 0; ci < 16; ++ci)
      s += ws[OFF_RW + i * 16 + ci] * inv512 * T0[1 * 256 + ci * 16 + co];
    ws[OFF_A0 + tid] = s;
  }

  if (i < 512) {
    float ub[16], dg[16], rm[16], cm[16], bv[16];
    #pragma unroll
    for (int ci = 0; ci < 16; ++ci) {
      ub[ci] = ws[OFF_U + i * 16 + ci] * inv784;
      dg[ci] = ws[OFF_DIAG + i * 16 + ci];
      rm[ci] = ws[OFF_ROWW + i * 16 + ci] * inv512;
      cm[ci] = ws[OFF_COLW + i * 16 + ci] * inv512;
      bv[ci] = bvec[i * 16 + ci];
    }
    float sB0 = 0.f, sR = 0.f, sC = 0.f, sD = 0.f, s2 = 0.f;
    #pragma unroll
    for (int ci = 0; ci < 16; ++ci) {
      int o = ci * 16 + co;
      sB0 += ub[ci] * T0[2 * 256 + o] + dg[ci] * T0[4 * 256 + o] +
             rm[ci] * T0[6 * 256 + o] + cm[ci] * T0[7 * 256 + o] +
             bv[ci] * T0[9 * 256 + o];
      sR  += ub[ci] * T1[3 * 256 + o] + dg[ci] * T1[8 * 256 + o] +
             rm[ci] * T1[17 * 256 + o] + cm[ci] * T1[18 * 256 + o] +
             bv[ci] * T1[23 * 256 + o];
      sC  += ub[ci] * T1[1 * 256 + o] + dg[ci] * T1[6 * 256 + o] +
             rm[ci] * T1[12 * 256 + o] + cm[ci] * T1[15 * 256 + o] +
             bv[ci] * T1[21 * 256 + o];
      sD  += ub[ci] * T1[0 * 256 + o] + dg[ci] * T1[5 * 256 + o] +
             rm[ci] * T1[10 * 256 + o] + cm[ci] * T1[14 * 256 + o] +
             bv[ci] * T1[20 * 256 + o];
      s2  += ub[ci] * T2[0 * 256 + o] + dg[ci] * T2[2 * 256 + o] +
             rm[ci] * T2[4 * 256 + o] + cm[ci] * T2[5 * 256 + o] +
             bv[ci] * T2[7 * 256 + o];
    }
    float cD = 0.f, c2 = 0.f;
    #pragma unroll
    for (int ci = 0; ci < 16; ++ci) {
      int o = ci * 16 + co;
      float mw = ws[OFF_MWIH + ci], ma = ws[OFF_MALL + ci];
      float md = ws[OFF_MDIAG + ci], mb = ws[OFF_MB + ci];
      cD += mw * T1[2 * 256 + o] + md * T1[7 * 256 + o] +
            ma * T1[16 * 256 + o] + mb * T1[22 * 256 + o];
      c2 += mw * T2[1 * 256 + o] + md * T2[3 * 256 + o] +
            ma * T2[6 * 256 + o] + mb * T2[8 * 256 + o];
    }
    ws[OFF_B0 + i * 16 + co]    = sB0;
    ws[OFF_ROW1 + i * 16 + co]  = sR;
    ws[OFF_COL1 + i * 16 + co]  = sC;
    ws[OFF_DIAG1 + i * 16 + co] = sD + cD;
    out2[i * 16 + co] = bias2[co] + s2 + c2;
  }

  if (i == 0) {
    float c0 = bias0[co], c1 = bias1[co];
    #pragma unroll
    for (int ci = 0; ci < 16; ++ci) {
      int o = ci * 16 + co;
      float mw = ws[OFF_MWIH + ci], ma = ws[OFF_MALL + ci];
      float md = ws[OFF_MDIAG + ci], mb = ws[OFF_MB + ci];
      c0 += mw * T0[3 * 256 + o] + md * T0[5 * 256 + o] +
            ma * T0[8 * 256 + o] + mb * T0[10 * 256 + o];
      c1 += mw * T1[4 * 256 + o] + md * T1[9 * 256 + o] +
            ma * T1[19 * 256 + o] + mb * T1[24 * 256 + o];
    }
    ws[OFF_C0 + co] = c0;
    ws[OFF_C1 + co] = c1;
  }
}

// ---------------- phase 3: WMMA main kernels ----------------
// A 16x4 f32 layout: lane L (M = L&15): VGPR0/1 hold K = {2*(L>>4), 2*(L>>4)+1} of chunk.
// B 4x16 f32 layout: lane L (N = L&15): VGPR0/1 hold K rows {2*(L>>4), 2*(L>>4)+1}.
// C/D 16x16 f32: VGPR v, lane L -> row M = v + 8*(L>>4), col N = L&15.

__global__ __launch_bounds__(32) void k_out0(const float* __restrict__ wih,
                                             const float* __restrict__ T0,
                                             const float* __restrict__ A0,
                                             const float* __restrict__ B0,
                                             const float* __restrict__ C0,
                                             float* __restrict__ out0) {
  int blk = blockIdx.x;
  int a  = blk >> 5;
  int b0 = (blk & 31) << 4;
  int lane = threadIdx.x;
  int M = lane & 15, half = lane >> 4;

  const float* src = wih + ((size_t)a * 512u + b0 + M) * 16u;
  v8f acc = {};
  #pragma unroll
  for (int k = 0; k < 4; ++k) {
    int ch = 4 * k + 2 * half;
    v2f Aa; Aa.x = src[ch];            Aa.y = src[ch + 1];
    v2f Bb; Bb.x = T0[ch * 16 + M];    Bb.y = T0[(ch + 1) * 16 + M];   // theta_0[0]
    acc = __builtin_amdgcn_wmma_f32_16x16x4_f32(false, Aa, false, Bb,
                                                (short)0, acc, false, false);
  }
  float aAdd = A0[a * 16 + M];
  float cAdd = C0[M];
  #pragma unroll
  for (int v = 0; v < 8; ++v) {
    int brow = b0 + v + 8 * half;
    out0[((size_t)a * 512u + brow) * 16u + M] =
        acc[v] + aAdd + B0[brow * 16 + M] + cAdd;
  }
}

__global__ __launch_bounds__(32) void k_out1(const float* __restrict__ whh,
                                             const float* __restrict__ T1,
                                             const float* __restrict__ Row1,
                                             const float* __restrict__ Col1,
                                             const float* __restrict__ Diag1,
                                             const float* __restrict__ C1,
                                             float* __restrict__ out1) {
  int blk = blockIdx.x;
  int r  = blk >> 5;
  int s0 = (blk & 31) << 4;
  int lane = threadIdx.x;
  int M = lane & 15, half = lane >> 4;

  const float* T13 = T1 + 13 * 256;   // identity W[r,s]
  const float* T11 = T1 + 11 * 256;   // transpose W[s,r]
  const float* srcI = whh + ((size_t)r * 512u + s0 + M) * 16u;
  const float* srcT = whh + ((size_t)(s0 + M) * 512u + r) * 16u;

  v8f acc = {};
  #pragma unroll
  for (int k = 0; k < 4; ++k) {
    int ch = 4 * k + 2 * half;
    v2f Ai; Ai.x = srcI[ch];            Ai.y = srcI[ch + 1];
    v2f Bi; Bi.x = T13[ch * 16 + M];    Bi.y = T13[(ch + 1) * 16 + M];
    acc = __builtin_amdgcn_wmma_f32_16x16x4_f32(false, Ai, false, Bi,
                                                (short)0, acc, false, false);
    v2f At; At.x = srcT[ch];            At.y = srcT[ch + 1];
    v2f Bt; Bt.x = T11[ch * 16 + M];    Bt.y = T11[(ch + 1) * 16 + M];
    acc = __builtin_amdgcn_wmma_f32_16x16x4_f32(false, At, false, Bt,
                                                (short)0, acc, false, false);
  }
  float rAdd = Row1[r * 16 + M];
  float cAdd = C1[M];
  float dAdd = Diag1[r * 16 + M];
  #pragma unroll
  for (int v = 0; v < 8; ++v) {
    int s = s0 + v + 8 * half;
    float val = acc[v] + rAdd + Col1[s * 16 + M] + cAdd;
    if (s == r) val += dAdd;
    out1[((size_t)r * 512u + s) * 16u + M] = val;
  }
}

// ---------------- host ----------------

extern "C" void kernel_launch(void* const* d_in, const int* in_sizes, int n_in,
                              void* d_out, int out_size, void* d_ws, size_t ws_size,
                              hipStream_t stream) {
  const float* wih   = (const float*)d_in[0];   // (784,512,16)
  const float* whh   = (const float*)d_in[1];   // (512,512,16)
  const float* bvec  = (const float*)d_in[2];   // (512,16)
  const float* T0    = (const float*)d_in[3];   // (11,16,16)
  const float* bias0 = (const float*)d_in[4];   // (16)
  const float* T1    = (const float*)d_in[5];   // (25,16,16)
  const float* bias1 = (const float*)d_in[6];   // (16)
  const float* T2    = (const float*)d_in[7];   // (9,16,16)
  const float* bias2 = (const float*)d_in[8];   // (16)

  float* out  = (float*)d_out;
  float* out0 = out;                       // 784*512*16 = 6422528
  float* out1 = out + 6422528;             // 512*512*16 = 4194304
  float* out2 = out + 10616832;            // 512*16     = 8192
  float* ws   = (float*)d_ws;

  k_zero  <<<(45376 + 255) / 256, 256, 0, stream>>>(ws, 45376);
  k_sum_u <<<256, 256, 0, stream>>>(wih, ws + OFF_U);
  k_sum_rw<<<(8 * 12544 + 255) / 256, 256, 0, stream>>>(wih, ws + OFF_RW);
  k_diag  <<<32, 256, 0, stream>>>(whh, ws + OFF_DIAG);
  k_roww  <<<256, 256, 0, stream>>>(whh, ws + OFF_ROWW);
  k_colw  <<<256, 256, 0, stream>>>(whh, ws + OFF_COLW);
  k_small <<<1, 16, 0, stream>>>(ws, bvec);
  k_combine<<<49, 256, 0, stream>>>(ws, bvec, T0, bias0, T1, bias1, T2, bias2, out2);
  k_out0  <<<784 * 32, 32, 0, stream>>>(wih, T0, ws + OFF_A0, ws + OFF_B0, ws + OFF_C0, out0);
  k_out1  <<<512 * 32, 32, 0, stream>>>(whh, T1, ws + OFF_ROW1, ws + OFF_COL1, ws + OFF_DIAG1, ws + OFF_C1, out1);
}